// RPN_20083267076517
// MI455X (gfx1250) — compile-verified
//
#include <hip/hip_runtime.h>
#include <math.h>

#define Bn 4
#define Cn 512
#define Hn 64
#define Wn 64
#define OCn 512
#define KC 16
#define KCH 8          // KC/2 channel pairs
#define OCT 64
#define TOPN 500

typedef float v2f_t __attribute__((ext_vector_type(2)));
typedef float v8f_t __attribute__((ext_vector_type(8)));

// ---------------- kernel 1: weight transpose [oc][ic][ky][kx] -> [tap][ic][oc] ----
__global__ __launch_bounds__(256) void k_wt(const float* __restrict__ w,
                                            float* __restrict__ wt) {
  int i = blockIdx.x * 256 + threadIdx.x;
  const int total = OCn * Cn * 9;
  if (i >= total) return;
  int tap = i % 9;
  int ic  = (i / 9) % Cn;
  int oc  = i / (9 * Cn);
  wt[(tap * Cn + ic) * OCn + oc] = w[i];
}

// ---------------- kernel 2: 3x3 conv + bias + ReLU via V_WMMA_F32_16X16X4_F32 ----
// Block: 256 threads (8 waves) -> 64 oc x 64 x x 2 rows.
// Wave (w): ocsub2=w&1 (32-oc half), xsub=(w>>1)&1 (32-x half), yrow=w>>2 (row).
// Each wave: 4 accumulators = 32(oc) x 32(x) of one row.
// LDS holds channel PAIRS contiguously so each WMMA fragment is one ds_load_b64.
__global__ __launch_bounds__(256) void k_conv(const float* __restrict__ in,
                                              const float* __restrict__ wt,
                                              const float* __restrict__ bias,
                                              float* __restrict__ f) {
  __shared__ float lds_in[4][KCH][68][2];   // rows y0-1..y0+2, chan-pair, x halo, pair
  __shared__ float lds_w[9][KCH][OCT][2];   // tap, chan-pair, oc, pair

  const int blk = blockIdx.x;               // b*32 + y0/2
  const int bb = blk >> 5;
  const int y0 = (blk & 31) * 2;
  const int ocbase = blockIdx.y * OCT;

  const int tid  = threadIdx.x;
  const int wave = tid >> 5;
  const int lane = tid & 31;
  const int ocsub2 = wave & 1;              // oc base ocsub2*32
  const int xsub   = (wave >> 1) & 1;       // x base xsub*32
  const int yrow   = wave >> 2;             // output row y0 + yrow

  const int ln15  = lane & 15;
  const int khalf = (lane >> 4) << 1;       // 0 (lanes 0-15) or 2 (lanes 16-31)

  v8f_t acc0 = {0.f, 0.f, 0.f, 0.f, 0.f, 0.f, 0.f, 0.f};  // (oc+0 , x+0 )
  v8f_t acc1 = acc0;                                      // (oc+0 , x+16)
  v8f_t acc2 = acc0;                                      // (oc+16, x+0 )
  v8f_t acc3 = acc0;                                      // (oc+16, x+16)

  for (int kc = 0; kc < Cn; kc += KC) {
    __syncthreads();
    // stage input: (r, cpair, xh), channel pair packed; zero-padded halo
    for (int i = tid; i < 4 * KCH * 66; i += 256) {
      int xh = i % 66;
      int rc = i / 66;
      int cp = rc % KCH;
      int r  = rc / KCH;
      int gy = y0 + r - 1;
      int gx = xh - 1;
      float v0 = 0.f, v1 = 0.f;
      if ((unsigned)gy < (unsigned)Hn && (unsigned)gx < (unsigned)Wn) {
        const float* p = in + ((size_t)(bb * Cn + kc + 2 * cp) * Hn + gy) * Wn + gx;
        v0 = p[0];
        v1 = p[Hn * Wn];
      }
      lds_in[r][cp][xh][0] = v0;
      lds_in[r][cp][xh][1] = v1;
    }
    // stage weights: (tap, cpair, oc), coalesced along oc per global read
    for (int i = tid; i < 9 * KCH * OCT; i += 256) {
      int oc  = i & (OCT - 1);
      int tc  = i >> 6;
      int cp  = tc % KCH;
      int tap = tc / KCH;
      const float* p = wt + (size_t)(tap * Cn + kc + 2 * cp) * OCn + ocbase + oc;
      lds_w[tap][cp][oc][0] = p[0];
      lds_w[tap][cp][oc][1] = p[OCn];
    }
    __syncthreads();

#pragma unroll
    for (int ky = 0; ky < 3; ++ky) {
#pragma unroll
      for (int kx = 0; kx < 3; ++kx) {
        const int tap = ky * 3 + kx;
#pragma unroll
        for (int k4 = 0; k4 < KC; k4 += 4) {
          const int cp = (k4 + khalf) >> 1;   // channel pair index for this lane half
          // A fragments: 16(oc) x 4(c); one b64 each
          v2f_t a0 = *(const v2f_t*)&lds_w[tap][cp][ocsub2 * 32 + ln15][0];
          v2f_t a1 = *(const v2f_t*)&lds_w[tap][cp][ocsub2 * 32 + 16 + ln15][0];
          // B fragments: 4(c) x 16(x); one b64 each
          const int xi = xsub * 32 + ln15 + kx;
          v2f_t b0 = *(const v2f_t*)&lds_in[yrow + ky][cp][xi][0];
          v2f_t b1 = *(const v2f_t*)&lds_in[yrow + ky][cp][xi + 16][0];
          acc0 = __builtin_amdgcn_wmma_f32_16x16x4_f32(
              false, a0, false, b0, (short)0, acc0, false, false);
          acc1 = __builtin_amdgcn_wmma_f32_16x16x4_f32(
              false, a0, false, b1, (short)0, acc1, false, false);
          acc2 = __builtin_amdgcn_wmma_f32_16x16x4_f32(
              false, a1, false, b0, (short)0, acc2, false, false);
          acc3 = __builtin_amdgcn_wmma_f32_16x16x4_f32(
              false, a1, false, b1, (short)0, acc3, false, false);
        }
      }
    }
  }

  // D layout: VGPR v, lanes 0-15 -> M=v, N=lane; lanes 16-31 -> M=v+8, N=lane-16
  const int y   = y0 + yrow;
  const int Mhi = (lane & 16) ? 8 : 0;
  const int x0  = xsub * 32 + ln15;
#pragma unroll
  for (int v = 0; v < 8; ++v) {
    int ocA = ocbase + ocsub2 * 32 + v + Mhi;        // acc0/acc1 oc
    int ocB = ocA + 16;                              // acc2/acc3 oc
    float bvA = bias[ocA];
    float bvB = bias[ocB];
    float r0 = acc0[v] + bvA; r0 = r0 > 0.f ? r0 : 0.f;
    float r1 = acc1[v] + bvA; r1 = r1 > 0.f ? r1 : 0.f;
    float r2 = acc2[v] + bvB; r2 = r2 > 0.f ? r2 : 0.f;
    float r3 = acc3[v] + bvB; r3 = r3 > 0.f ? r3 : 0.f;
    f[((size_t)(bb * OCn + ocA) * Hn + y) * Wn + x0]      = r0;
    f[((size_t)(bb * OCn + ocA) * Hn + y) * Wn + x0 + 16] = r1;
    f[((size_t)(bb * OCn + ocB) * Hn + y) * Wn + x0]      = r2;
    f[((size_t)(bb * OCn + ocB) * Hn + y) * Wn + x0 + 16] = r3;
  }
}

// ---------------- kernel 3: 1x1 heads + anchor decode ----------------
__global__ __launch_bounds__(320) void k_heads(const float* __restrict__ f,
                                               const float* __restrict__ sw,
                                               const float* __restrict__ sb,
                                               const float* __restrict__ bw,
                                               const float* __restrict__ bbias,
                                               float* __restrict__ logits,
                                               float* __restrict__ props) {
  const int by = blockIdx.x;   // b*64 + y
  const int bb = by >> 6;
  const int y  = by & 63;
  const int tid = threadIdx.x;
  const int o = tid >> 6;      // 0 = score, 1..4 = bbox channels
  const int x = tid & 63;
  __shared__ float reg4[4][64];

  const float* wv = (o == 0) ? sw : (bw + (o - 1) * Cn);
  const float* fp = f + ((size_t)(bb * Cn) * Hn + y) * Wn + x;
  float acc = 0.f;
  for (int c = 0; c < Cn; ++c)
    acc += fp[(size_t)c * Hn * Wn] * wv[c];

  if (o == 0) {
    logits[bb * 4096 + y * 64 + x] = acc + sb[0];
  } else {
    reg4[o - 1][x] = acc + bbias[o - 1];
  }
  __syncthreads();
  if (o == 0) {
    float dx = reg4[0][x], dy = reg4[1][x], dw = reg4[2][x], dh = reg4[3][x];
    // anchors: size 32, stride 800//64 = 12, centered at (12x, 12y), wa=ha=32
    float cx = dx * 32.f + (float)x * 12.f;
    float cy = dy * 32.f + (float)y * 12.f;
    float w2 = 0.5f * expf(dw) * 32.f;
    float h2 = 0.5f * expf(dh) * 32.f;
    int n = bb * 4096 + y * 64 + x;
    props[n * 4 + 0] = cx - w2;
    props[n * 4 + 1] = cy - h2;
    props[n * 4 + 2] = cx + w2;
    props[n * 4 + 3] = cy + h2;
  }
}

// ---------------- kernel 4: per-batch bitonic top-k + greedy NMS ----------------
__global__ __launch_bounds__(1024) void k_sortnms(const float* __restrict__ logits,
                                                  const float* __restrict__ props,
                                                  float* __restrict__ out) {
  __shared__ float vals[4096];
  __shared__ int   idxA[4096];
  __shared__ float bx[TOPN][4];
  __shared__ float sc[TOPN];
  __shared__ float ar[TOPN];
  __shared__ int   kp[TOPN];

  const int bb = blockIdx.x;
  const int tid = threadIdx.x;

  for (int i = tid; i < 4096; i += 1024) {
    vals[i] = logits[bb * 4096 + i];
    idxA[i] = i;
  }
  __syncthreads();

  // bitonic sort: descending by value, ties -> lower index first (matches top_k)
  for (unsigned k = 2; k <= 4096u; k <<= 1) {
    for (unsigned j = k >> 1; j > 0; j >>= 1) {
      for (unsigned s = 0; s < 4; ++s) {
        unsigned i = tid + s * 1024;
        unsigned l = i ^ j;
        if (l > i) {
          float vi = vals[i], vl = vals[l];
          int ii = idxA[i], il = idxA[l];
          bool iFirst  = (vi > vl) || (vi == vl && ii < il);
          bool descSeg = ((i & k) == 0);
          bool doSwap  = descSeg ? !iFirst : iFirst;
          if (doSwap) { vals[i] = vl; vals[l] = vi; idxA[i] = il; idxA[l] = ii; }
        }
      }
      __syncthreads();
    }
  }

  for (int t = tid; t < TOPN; t += 1024) {
    int n = idxA[t];
    const float* p = props + ((size_t)(bb * 4096 + n)) * 4;
    float x1 = fminf(fmaxf(p[0], 0.f), 800.f);
    float y1 = fminf(fmaxf(p[1], 0.f), 800.f);
    float x2 = fminf(fmaxf(p[2], 0.f), 800.f);
    float y2 = fminf(fmaxf(p[3], 0.f), 800.f);
    bx[t][0] = x1; bx[t][1] = y1; bx[t][2] = x2; bx[t][3] = y2;
    float pr = 1.f / (1.f + expf(-vals[t]));
    sc[t] = pr;
    ar[t] = (x2 - x1) * (y2 - y1);
    kp[t] = (((x2 - x1) >= 0.001f) && ((y2 - y1) >= 0.001f) && (pr >= 0.f)) ? 1 : 0;
  }
  __syncthreads();

  for (int i = 0; i < TOPN; ++i) {
    if (kp[i]) {                          // uniform branch (LDS value, after barrier)
      float xi1 = bx[i][0], yi1 = bx[i][1], xi2 = bx[i][2], yi2 = bx[i][3], ai = ar[i];
      for (int t = tid; t < TOPN; t += 1024) {
        if (t > i && kp[t]) {
          float lx = fmaxf(xi1, bx[t][0]);
          float ly = fmaxf(yi1, bx[t][1]);
          float rx = fminf(xi2, bx[t][2]);
          float ry = fminf(yi2, bx[t][3]);
          float iw = fmaxf(rx - lx, 0.f);
          float ih = fmaxf(ry - ly, 0.f);
          float inter = iw * ih;
          float iou = inter / (ai + ar[t] - inter + 1e-9f);
          if (iou > 0.7f) kp[t] = 0;
        }
      }
    }
    __syncthreads();
  }

  const size_t F_ELEMS = (size_t)Bn * Cn * Hn * Wn;   // 8388608
  float* oprops  = out + F_ELEMS;
  float* oscores = oprops + (size_t)Bn * TOPN * 4;
  float* okeep   = oscores + (size_t)Bn * TOPN;
  for (int t = tid; t < TOPN; t += 1024) {
    float kf = (float)kp[t];
    oprops[((size_t)(bb * TOPN + t)) * 4 + 0] = bx[t][0] * kf;
    oprops[((size_t)(bb * TOPN + t)) * 4 + 1] = bx[t][1] * kf;
    oprops[((size_t)(bb * TOPN + t)) * 4 + 2] = bx[t][2] * kf;
    oprops[((size_t)(bb * TOPN + t)) * 4 + 3] = bx[t][3] * kf;
    oscores[bb * TOPN + t] = sc[t] * kf;
    okeep[bb * TOPN + t]   = kf;
  }
}

extern "C" void kernel_launch(void* const* d_in, const int* in_sizes, int n_in,
                              void* d_out, int out_size, void* d_ws, size_t ws_size,
                              hipStream_t stream) {
  const float* feat  = (const float*)d_in[0];
  const float* convw = (const float*)d_in[1];
  const float* convb = (const float*)d_in[2];
  const float* sw    = (const float*)d_in[3];
  const float* sb    = (const float*)d_in[4];
  const float* bw    = (const float*)d_in[5];
  const float* bbias = (const float*)d_in[6];
  float* out = (float*)d_out;

  char* ws = (char*)d_ws;
  float* wt     = (float*)ws;                              // 9*512*512 f32 = 9437184 B
  float* logits = (float*)(ws + 9437184);                  // 4*4096 f32    = 65536 B
  float* props  = (float*)(ws + 9437184 + 65536);          // 4*4096*4 f32  = 262144 B

  k_wt<<<(OCn * Cn * 9 + 255) / 256, 256, 0, stream>>>(convw, wt);
  k_conv<<<dim3(Bn * Hn / 2, OCn / OCT), 256, 0, stream>>>(feat, wt, convb, out);
  k_heads<<<Bn * Hn, 320, 0, stream>>>(out, sw, sb, bw, bbias, logits, props);
  k_sortnms<<<Bn, 1024, 0, stream>>>(logits, props, out);
}